// SupConLoss_45715631898733
// MI455X (gfx1250) — compile-verified
//
#include <hip/hip_runtime.h>
#include <hip/hip_bf16.h>
#include <math.h>

// ---------------------------------------------------------------------------
// SupConLoss fused implementation for MI455X (gfx1250, wave32, WMMA).
//
// Heavy compute: 4096x4096x256 fp32 Gram matrix via V_WMMA_F32_16X16X4_F32,
// fused with a constant-shift log-sum-exp (shift = 1/T, the provable row max
// for unit vectors) so no rowmax pass / online softmax is needed.
// Positive-pair quantile logic collapses to a tiny per-row selection
// (~2m-1 ~ 40 values), handled by a separate cheap kernel.
//
// Gram kernel: each wave processes TWO 16-col tiles concurrently (two
// independent WMMA accumulation chains, A LDS loads reused x2).
// ---------------------------------------------------------------------------

typedef __attribute__((ext_vector_type(2))) float v2f;
typedef __attribute__((ext_vector_type(8))) float v8f;

#define BSZ        2048
#define NVIEWS     2
#define NROWS      4096          // NVIEWS * BSZ
#define DIM        256
#define NCLASSES   100
#define INV_T      14.2857142857142857f   // 1 / 0.07

// ---------------------------------------------------------------------------
// Kernel 1: L2-normalize each (b, v) embedding, write view-major cf [N, 256].
// grid = 4096 (one vector per block), block = 256.
// ---------------------------------------------------------------------------
__global__ __launch_bounds__(256) void sc_normalize_kernel(
    const float* __restrict__ feats, float* __restrict__ cf) {
  __shared__ float red[256];
  const int vec = blockIdx.x;            // vec = b*2 + v  (feats flat layout)
  const int b = vec >> 1;
  const int v = vec & 1;
  const float x = feats[(size_t)vec * DIM + threadIdx.x];
  red[threadIdx.x] = x * x;
  __syncthreads();
  for (int off = 128; off > 0; off >>= 1) {
    if (threadIdx.x < (unsigned)off) red[threadIdx.x] += red[threadIdx.x + off];
    __syncthreads();
  }
  const float nrm = sqrtf(red[0]);
  cf[((size_t)(v * BSZ + b)) * DIM + threadIdx.x] = x / nrm;
}

// ---------------------------------------------------------------------------
// Kernel 2: deterministic per-class member lists (ascending order).
// launch <<<1,128>>>; thread c handles class c.
// ---------------------------------------------------------------------------
__global__ void sc_members_kernel(const int* __restrict__ labels,
                                  int* __restrict__ counts,
                                  int* __restrict__ members) {
  const int c = threadIdx.x;
  if (c >= NCLASSES) return;
  int cnt = 0;
  for (int bb = 0; bb < BSZ; ++bb)
    if (labels[bb] == c) members[c * BSZ + cnt++] = bb;
  counts[c] = cnt;
}

// ---------------------------------------------------------------------------
// Kernel 3: Gram matrix + fused exp-sum (denominator of the softmax).
//   denom[i] = sum_{j != i} exp(dot(cf_i, cf_j)/T - 1/T)
// grid = 256 blocks (16 rows each), block = 256 threads = 8 waves.
// A-panel (16 x 256 f32) staged in LDS; waves split the 256 column tiles,
// two tiles per wave iteration (dual accumulators, shared A loads).
// ---------------------------------------------------------------------------
__global__ __launch_bounds__(256) void sc_gram_expsum_kernel(
    const float* __restrict__ cf, float* __restrict__ denom) {
  __shared__ float As[16 * DIM];          // 16 KB A panel
  __shared__ float partS[8][32][8];       // 8 KB partial sums

  const int rowTile = blockIdx.x;         // 0..255
  const int rowbase = rowTile * 16;

  for (int idx = threadIdx.x; idx < 16 * DIM; idx += 256)
    As[idx] = cf[(size_t)rowbase * DIM + idx];
  __syncthreads();

  const int wave = threadIdx.x >> 5;      // wave32
  const int lane = threadIdx.x & 31;
  const int nsub = lane & 15;             // M (for A) / N (for B) index
  const int ksub = (lane >> 4) * 2;       // K sub-offset (half-wave split)
  const int rsub = (lane >> 4) * 8;       // row offset of accumulator regs

  float s[8];
#pragma unroll
  for (int v = 0; v < 8; ++v) s[v] = 0.f;

  const float* aptr = &As[nsub * DIM + ksub];

  for (int ct0 = wave * 2; ct0 < NROWS / 16; ct0 += 16) {
    const int ct1 = ct0 + 1;
    const float* bptr0 = cf + (size_t)(ct0 * 16 + nsub) * DIM + ksub;
    const float* bptr1 = cf + (size_t)(ct1 * 16 + nsub) * DIM + ksub;
    // Prefetch the next pair of column tiles this wave will touch.
    __builtin_prefetch(cf + (size_t)((ct0 + 16) * 16 + nsub) * DIM, 0, 1);

    v8f c0 = {0.f, 0.f, 0.f, 0.f, 0.f, 0.f, 0.f, 0.f};
    v8f c1 = {0.f, 0.f, 0.f, 0.f, 0.f, 0.f, 0.f, 0.f};
#pragma unroll 8
    for (int kk = 0; kk < DIM; kk += 4) {
      v2f a  = *(const v2f*)(aptr + kk);    // ds_load from LDS A panel
      v2f b0 = *(const v2f*)(bptr0 + kk);   // global_load_b64 (L2-resident)
      v2f b1 = *(const v2f*)(bptr1 + kk);
      c0 = __builtin_amdgcn_wmma_f32_16x16x4_f32(
               false, a, false, b0, (short)0, c0, false, false);
      c1 = __builtin_amdgcn_wmma_f32_16x16x4_f32(
               false, a, false, b1, (short)0, c1, false, false);
    }

    const bool diag0 = (ct0 == rowTile);  // only tiles that can hold j == i
    const bool diag1 = (ct1 == rowTile);
#pragma unroll
    for (int v = 0; v < 8; ++v) {
      const bool self = (nsub == rsub + v);
      const float x0 = c0[v] * INV_T;     // adc value, tile 0
      const float e0 = __expf(x0 - INV_T);
      s[v] += (diag0 && self) ? 0.f : e0;
      const float x1 = c1[v] * INV_T;     // adc value, tile 1
      const float e1 = __expf(x1 - INV_T);
      s[v] += (diag1 && self) ? 0.f : e1;
    }
  }

#pragma unroll
  for (int v = 0; v < 8; ++v) partS[wave][lane][v] = s[v];
  __syncthreads();

  if (threadIdx.x < 16) {                 // one thread per row of the strip
    const int r = threadIdx.x;
    const int half = r >> 3;              // which lane half holds row r
    const int vv = r & 7;                 // which accumulator register
    float tot = 0.f;
    for (int w = 0; w < 8; ++w)
      for (int l = 0; l < 16; ++l)
        tot += partS[w][half * 16 + l][vv];
    denom[rowbase + r] = tot;
  }
}

// ---------------------------------------------------------------------------
// Kernel 4: per-row positive-pair quantile mask + masked sum/count.
// Row i positives: columns j with same label, j != i (P = 2m-1 values).
// temp = (dot + 2)/T > 0 on positives; sorted-row quantile reduces to a
// rank-select among the positives (window = 2 zeros + P positives).
// grid = 4096, block = 128.
// ---------------------------------------------------------------------------
__global__ __launch_bounds__(128) void sc_positives_kernel(
    const float* __restrict__ cf, const int* __restrict__ labels,
    const int* __restrict__ counts, const int* __restrict__ members,
    const int* __restrict__ threshp,
    float* __restrict__ psum, float* __restrict__ pcnt) {
  __shared__ float vals[NROWS];           // worst case P = 4095 (16 KB)
  __shared__ int   sh_selfpos;
  __shared__ float sh_quant;
  __shared__ float redS[128];
  __shared__ float redC[128];

  const int i      = blockIdx.x;
  const int b_self = i & (BSZ - 1);
  const int v_self = i >> 11;
  const int lab    = labels[b_self];
  const int m      = counts[lab];
  const int off    = lab * BSZ;
  const int P      = 2 * m - 1;

  for (int s = threadIdx.x; s < m; s += blockDim.x)
    if (members[off + s] == b_self) sh_selfpos = s;
  if (threadIdx.x == 0) sh_quant = 0.f;
  __syncthreads();
  const int t_self = 2 * sh_selfpos + v_self;

  const float* ri = cf + (size_t)i * DIM;
  for (int t = threadIdx.x; t < 2 * m; t += blockDim.x) {
    if (t == t_self) continue;
    const int s  = t >> 1;
    const int vv = t & 1;
    const int j  = vv * BSZ + members[off + s];
    const float* rj = cf + (size_t)j * DIM;
    float dot = 0.f;
#pragma unroll 4
    for (int k = 0; k < DIM; ++k) dot = fmaf(ri[k], rj[k], dot);
    const int p = t - (t > t_self ? 1 : 0);   // compact out the self slot
    vals[p] = (dot + 2.0f) * INV_T;           // temp value
  }
  __syncthreads();

  // quantile index in the length-(2m+1) window: idx = ceil(q*(L-1)).
  const float q  = 1.0f - (float)threshp[0];
  const int  Lw  = 2 * m + 1;
  const int idxq = (int)ceilf(q * (float)(Lw - 1));
  const int kq   = idxq - 2;                  // rank among positives; <0 -> 0

  if (kq >= 0) {
    for (int p = threadIdx.x; p < P; p += blockDim.x) {
      const float vp = vals[p];
      int rank = 0;
      for (int s2 = 0; s2 < P; ++s2) {
        const float vs = vals[s2];
        rank += (vs < vp) || (vs == vp && s2 < p);   // stable rank
      }
      if (rank == kq) sh_quant = vp;
    }
  }
  __syncthreads();

  const float quant = sh_quant;
  float su = 0.f, cn = 0.f;
  for (int p = threadIdx.x; p < P; p += blockDim.x) {
    const float v2 = vals[p];
    if (v2 >= quant) { su += v2; cn += 1.f; }
  }
  redS[threadIdx.x] = su;
  redC[threadIdx.x] = cn;
  __syncthreads();
  for (int o = 64; o > 0; o >>= 1) {
    if (threadIdx.x < (unsigned)o) {
      redS[threadIdx.x] += redS[threadIdx.x + o];
      redC[threadIdx.x] += redC[threadIdx.x + o];
    }
    __syncthreads();
  }
  if (threadIdx.x == 0) {
    const float cnt = redC[0];
    // sum of adc over masked = sum(temp) - cnt * (2/T)
    psum[i] = redS[0] - cnt * (2.0f * INV_T);
    pcnt[i] = cnt;
  }
}

// ---------------------------------------------------------------------------
// Kernel 5: final reduction.
// mean_log_prob_pos_i = psum_i/cnt_i - C - log(denom_i)   (C = 1/T shift;
// shift-invariant identity replaces the reference's per-row max).
// loss = -(1/N) * sum_i mean_log_prob_pos_i     (T/BASE_T == 1)
// ---------------------------------------------------------------------------
__global__ __launch_bounds__(256) void sc_finalize_kernel(
    const float* __restrict__ denom, const float* __restrict__ psum,
    const float* __restrict__ pcnt, float* __restrict__ out) {
  __shared__ float red[256];
  float acc = 0.f;
  for (int r = threadIdx.x; r < NROWS; r += 256)
    acc += psum[r] / pcnt[r] - INV_T - logf(denom[r]);
  red[threadIdx.x] = acc;
  __syncthreads();
  for (int o = 128; o > 0; o >>= 1) {
    if (threadIdx.x < (unsigned)o) red[threadIdx.x] += red[threadIdx.x + o];
    __syncthreads();
  }
  if (threadIdx.x == 0) out[0] = -red[0] / (float)NROWS;
}

// ---------------------------------------------------------------------------
// Launcher.  Inputs: d_in[0]=features (2048*2*256 f32), d_in[1]=labels (2048
// i32), d_in[2]=thresh (1 i32).  Output: 1 f32 scalar.
// ---------------------------------------------------------------------------
extern "C" void kernel_launch(void* const* d_in, const int* in_sizes, int n_in,
                              void* d_out, int out_size, void* d_ws, size_t ws_size,
                              hipStream_t stream) {
  const float* feats  = (const float*)d_in[0];
  const int*   labels = (const int*)d_in[1];
  const int*   thresh = (const int*)d_in[2];
  float* out = (float*)d_out;

  // Workspace layout (~5.1 MB total).
  char* ws = (char*)d_ws;
  size_t o = 0;
  float* cf      = (float*)(ws + o); o += (size_t)NROWS * DIM * sizeof(float);
  float* denom   = (float*)(ws + o); o += (size_t)NROWS * sizeof(float);
  float* psum    = (float*)(ws + o); o += (size_t)NROWS * sizeof(float);
  float* pcnt    = (float*)(ws + o); o += (size_t)NROWS * sizeof(float);
  int*   counts  = (int*)  (ws + o); o += 128 * sizeof(int);
  int*   members = (int*)  (ws + o); o += (size_t)NCLASSES * BSZ * sizeof(int);
  (void)ws_size; (void)in_sizes; (void)n_in; (void)out_size;

  sc_normalize_kernel<<<NROWS, 256, 0, stream>>>(feats, cf);
  sc_members_kernel<<<1, 128, 0, stream>>>(labels, counts, members);
  sc_gram_expsum_kernel<<<NROWS / 16, 256, 0, stream>>>(cf, denom);
  sc_positives_kernel<<<NROWS, 128, 0, stream>>>(cf, labels, counts, members,
                                                 thresh, psum, pcnt);
  sc_finalize_kernel<<<1, 256, 0, stream>>>(denom, psum, pcnt, out);
}